// FancyNet_59425167507641
// MI455X (gfx1250) — compile-verified
//
#include <hip/hip_runtime.h>

typedef __attribute__((ext_vector_type(2))) float v2f;
typedef __attribute__((ext_vector_type(8))) float v8f;

#define FN_D 20
#define TILE_M 16
#define LDS_STRIDE 34       // even (8B-aligned float2), >=32 (cols 0..31 fit), !=0 mod 8 (no bank conflicts)
#define WAVES_PER_BLOCK 8
#define TILES_PER_WAVE 4

static __device__ __forceinline__ v8f wmma_f32x4(v2f a, v2f b, v8f c) {
  // D = A(16x4 f32) * B(4x16 f32) + C(16x16 f32)
  return __builtin_amdgcn_wmma_f32_16x16x4_f32(
      /*neg_a=*/false, a, /*neg_b=*/false, b,
      /*c_mod=*/(short)0, c, /*reuse_a=*/false, /*reuse_b=*/false);
}

__global__ __launch_bounds__(256) void fancy_init(float* acc) {
  acc[0] = 0.0f;  // sum
  acc[1] = 0.0f;  // sum of squares
}

__global__ __launch_bounds__(256) void fancy_main(
    const float* __restrict__ X, const float* __restrict__ W,
    const float* __restrict__ b, const float* __restrict__ C,
    float* __restrict__ acc, int B) {
  __shared__ float lds[WAVES_PER_BLOCK * TILE_M * LDS_STRIDE];

  const int lane = threadIdx.x & 31;
  const int wave = threadIdx.x >> 5;
  const int hi   = lane >> 4;      // 0: lanes 0-15, 1: lanes 16-31
  const int l15  = lane & 15;
  const int ntiles    = (B + TILE_M - 1) / TILE_M;
  const int gridWaves = (int)gridDim.x * WAVES_PER_BLOCK;

  float* my = lds + wave * TILE_M * LDS_STRIDE;  // wave-private transpose tile

  // ---- loop-invariant per-wave setup -------------------------------------
  // B fragments: bw = W^T (k,n), bc = C (k,n); zero-pad n >= 20.
  const int n0  = l15;             // output col, N-tile 0 (always < 20)
  const int n1  = 16 + l15;        // output col, N-tile 1 (valid if < 20)
  const bool n1v = (n1 < FN_D);

  v2f bw[5][2], bc[5][2];
#pragma unroll
  for (int s = 0; s < 5; ++s) {
    const int k0 = 4 * s + 2 * hi;             // per ISA 32b-B layout
#pragma unroll
    for (int t = 0; t < 2; ++t) {
      const int n = 16 * t + l15;
      if (n < FN_D) {
        bw[s][t].x = W[n * FN_D + k0];         // W^T[k][n] = W[n][k]
        bw[s][t].y = W[n * FN_D + k0 + 1];
        bc[s][t].x = C[k0 * FN_D + n];
        bc[s][t].y = C[(k0 + 1) * FN_D + n];
      } else {
        bw[s][t].x = 0.f; bw[s][t].y = 0.f;
        bc[s][t].x = 0.f; bc[s][t].y = 0.f;
      }
    }
  }
  const float bias0 = b[n0];
  const float bias1 = n1v ? b[n1] : 0.f;

  float lsum = 0.0f, lsq = 0.0f;

  // ---- grid-stride over 16-row tiles -------------------------------------
  for (int tile = (int)blockIdx.x * WAVES_PER_BLOCK + wave; tile < ntiles;
       tile += gridWaves) {
    const int rowBase = tile * TILE_M;

    // A fragments of X: lane holds row (l15), K = k0..k0+1.
    // Out-of-range rows are clamped (rows are independent; masked at the sum).
    const int r  = rowBase + l15;
    const int rc = (r < B) ? r : (B - 1);
    v2f a[5];
#pragma unroll
    for (int s = 0; s < 5; ++s) {
      const int k0 = 4 * s + 2 * hi;
      a[s] = *(const v2f*)(X + (long)rc * FN_D + k0);
    }

    // GEMM 1: h1 = X @ W^T + b (K=20 in 5 steps, 2 N-tiles)
    v8f c0, c1;
#pragma unroll
    for (int v = 0; v < 8; ++v) { c0[v] = bias0; c1[v] = bias1; }
#pragma unroll
    for (int s = 0; s < 5; ++s) {
      c0 = wmma_f32x4(a[s], bw[s][0], c0);
      c1 = wmma_f32x4(a[s], bw[s][1], c1);
    }

    // Transpose h1 (D-layout -> A-layout) through wave-private LDS.
    // Cols 20..31 are written but never read (K-reads stop at 19).
#pragma unroll
    for (int v = 0; v < 8; ++v) {
      const int m = v + 8 * hi;                // C/D layout: row = v + 8*hi
      my[m * LDS_STRIDE + n0] = c0[v];
      my[m * LDS_STRIDE + n1] = c1[v];
    }
    asm volatile("" ::: "memory");             // DS pipe is in-order per wave
#pragma unroll
    for (int s = 0; s < 5; ++s) {
      const int k0 = 4 * s + 2 * hi;
      a[s] = *(const v2f*)(my + l15 * LDS_STRIDE + k0);
    }

    // GEMM 2: h2 = relu(h1 @ C + 1)
#pragma unroll
    for (int v = 0; v < 8; ++v) { c0[v] = 1.0f; c1[v] = 1.0f; }
#pragma unroll
    for (int s = 0; s < 5; ++s) {
      c0 = wmma_f32x4(a[s], bc[s][0], c0);
      c1 = wmma_f32x4(a[s], bc[s][1], c1);
    }
#pragma unroll
    for (int v = 0; v < 8; ++v) { c0[v] = fmaxf(c0[v], 0.f); c1[v] = fmaxf(c1[v], 0.f); }

    // Transpose h2 through LDS.
#pragma unroll
    for (int v = 0; v < 8; ++v) {
      const int m = v + 8 * hi;
      my[m * LDS_STRIDE + n0] = c0[v];
      my[m * LDS_STRIDE + n1] = c1[v];
    }
    asm volatile("" ::: "memory");
#pragma unroll
    for (int s = 0; s < 5; ++s) {
      const int k0 = 4 * s + 2 * hi;
      a[s] = *(const v2f*)(my + l15 * LDS_STRIDE + k0);
    }

    // GEMM 3: h3 = h2 @ W^T + b
#pragma unroll
    for (int v = 0; v < 8; ++v) { c0[v] = bias0; c1[v] = bias1; }
#pragma unroll
    for (int s = 0; s < 5; ++s) {
      c0 = wmma_f32x4(a[s], bw[s][0], c0);
      c1 = wmma_f32x4(a[s], bw[s][1], c1);
    }

    // Local sum / sum-of-squares over valid rows & cols (selects, no branches).
#pragma unroll
    for (int v = 0; v < 8; ++v) {
      const int m = rowBase + v + 8 * hi;
      const bool mv = (m < B);
      const float x = mv ? c0[v] : 0.f;
      const float y = (mv && n1v) ? c1[v] : 0.f;
      lsum += x + y;
      lsq  += x * x + y * y;
    }
  }

  // ---- wave32 reduction, then 2 fp32 atomics per wave --------------------
#pragma unroll
  for (int off = 16; off > 0; off >>= 1) {
    lsum += __shfl_xor(lsum, off, 32);
    lsq  += __shfl_xor(lsq,  off, 32);
  }
  if (lane == 0) {
    unsafeAtomicAdd(&acc[0], lsum);   // global_atomic_add_f32
    unsafeAtomicAdd(&acc[1], lsq);
  }
}

__global__ __launch_bounds__(64) void fancy_finalize(const float* acc, float* out) {
  if (threadIdx.x == 0) {
    const float sum  = acc[0];
    const float norm = sqrtf(acc[1]);
    // while norm > 1: /= 2  ->  halve ceil(log2(norm)) times
    float k = (norm > 1.0f) ? ceilf(log2f(norm)) : 0.0f;
    float scale = exp2f(-k);
    if (norm * scale < 0.8f) scale *= 10.0f;
    out[0] = sum * scale;
  }
}

extern "C" void kernel_launch(void* const* d_in, const int* in_sizes, int n_in,
                              void* d_out, int out_size, void* d_ws, size_t ws_size,
                              hipStream_t stream) {
  const float* X = (const float*)d_in[0];
  const float* W = (const float*)d_in[1];
  const float* b = (const float*)d_in[2];
  const float* C = (const float*)d_in[3];
  float* out = (float*)d_out;
  float* acc = (float*)d_ws;

  const int B = in_sizes[0] / FN_D;                 // 1048576
  const int ntiles = (B + TILE_M - 1) / TILE_M;     // 65536
  const int wavesNeeded = (ntiles + TILES_PER_WAVE - 1) / TILES_PER_WAVE;
  int blocks = (wavesNeeded + WAVES_PER_BLOCK - 1) / WAVES_PER_BLOCK;  // 2048
  if (blocks < 1) blocks = 1;

  fancy_init<<<1, 1, 0, stream>>>(acc);
  fancy_main<<<blocks, 256, 0, stream>>>(X, W, b, C, acc, B);
  fancy_finalize<<<1, 64, 0, stream>>>(acc, out);
}